// RGCN_74345883894621
// MI455X (gfx1250) — compile-verified
//
#include <hip/hip_runtime.h>
#include <cstddef>
#include <cstdint>

#define NPAPER 100000
#define NAUTHOR 50000
#define DFEAT 128
#define DOUT 64
#define LDSW 132  // 128 + 4 pad -> conflict-free ds_load_b64 across 32 lanes

typedef float v2f __attribute__((ext_vector_type(2)));
typedef float v8f __attribute__((ext_vector_type(8)));

// -------- float atomic max via single hardware integer atomic (deterministic) ----
__device__ __forceinline__ void atomic_max_f32(float* addr, float v) {
    int iv = __float_as_int(v);
    if (iv >= 0) {
        atomicMax((int*)addr, iv);
    } else {
        atomicMin((unsigned int*)addr, (unsigned int)iv);
    }
}

// -------- fill buffer with -inf (float4 vectorized) ----------------------------
__global__ void fill_neginf_kernel(float4* __restrict__ p, long n4) {
    long i = (long)blockIdx.x * blockDim.x + threadIdx.x;
    if (i < n4) {
        float ninf = __int_as_float((int)0xFF800000u);
        p[i] = make_float4(ninf, ninf, ninf, ninf);
    }
}

// -------- segment max: one wave per edge, 32 lanes x float4 = 128 features -----
__global__ void segmax_kernel(const float* __restrict__ x,
                              const int* __restrict__ src,
                              const int* __restrict__ dst,
                              float* __restrict__ agg, int nEdges) {
    long gid = (long)blockIdx.x * blockDim.x + threadIdx.x;
    int edge = (int)(gid >> 5);
    int lane = (int)(gid & 31);
    if (edge >= nEdges) return;
    int s = src[edge];
    int d = dst[edge];
    const float4 v = *(const float4*)(x + (long)s * DFEAT + lane * 4);
    float* o = agg + (long)d * DFEAT + lane * 4;
    atomic_max_f32(o + 0, v.x);
    atomic_max_f32(o + 1, v.y);
    atomic_max_f32(o + 2, v.z);
    atomic_max_f32(o + 3, v.w);
}

// -------- accumulate one (A, W) pair into the WMMA accumulator -----------------
// mode 0: stage A strip via GLOBAL_LOAD_ASYNC_TO_LDS_B128 (no clamp needed)
// mode 1: stage A strip via VGPRs with -inf -> 0 clamp (agg sources)
__device__ __forceinline__ void accum_src(
    const float* __restrict__ A, const float* __restrict__ W, int mode,
    float* __restrict__ Alds, int rowBase, int nRows, int N, int nBase,
    int tid, int nThreads, int lane, v8f& acc)
{
    __syncthreads();  // protect LDS reads of previous source

    if (mode == 0) {
        // CDNA5 async copy: global -> LDS with no VGPR round trip (ASYNCcnt)
        for (int i = tid; i < 16 * 32; i += nThreads) {
            int r  = i >> 5;
            int c4 = (i & 31) * 4;
            unsigned long long gsrc =
                (unsigned long long)(size_t)(A + (long)(rowBase + r) * DFEAT + c4);
            unsigned ldsoff = (unsigned)(size_t)(Alds + r * LDSW + c4);
            asm volatile("global_load_async_to_lds_b128 %0, %1, off"
                         :: "v"(ldsoff), "v"(gsrc)
                         : "memory");
        }
        asm volatile("s_wait_asynccnt 0x0" ::: "memory");
    } else {
        // cooperative, coalesced load of 16 x 128 fp32 strip (clamp -inf -> 0)
        for (int i = tid; i < 16 * 32; i += nThreads) {
            int r  = i >> 5;
            int c4 = (i & 31) * 4;
            int gr = rowBase + r;
            float4 v = make_float4(0.f, 0.f, 0.f, 0.f);
            if (gr < nRows) v = *(const float4*)(A + (long)gr * DFEAT + c4);
            if (__float_as_uint(v.x) == 0xFF800000u) v.x = 0.f;
            if (__float_as_uint(v.y) == 0xFF800000u) v.y = 0.f;
            if (__float_as_uint(v.z) == 0xFF800000u) v.z = 0.f;
            if (__float_as_uint(v.w) == 0xFF800000u) v.w = 0.f;
            float* s = Alds + r * LDSW + c4;
            s[0] = v.x; s[1] = v.y; s[2] = v.z; s[3] = v.w;
        }
    }
    __syncthreads();

    const int arow = lane & 15;   // A row held by this lane (ISA A-matrix layout)
    const int col  = lane & 15;   // B/C column held by this lane
    const int half = lane >> 4;   // K sub-pair selector
    const float* arp = Alds + arow * LDSW + half * 2;
    const float* wp  = W + (long)(half * 2) * N + nBase + col;

    #pragma unroll 8
    for (int k = 0; k < DFEAT; k += 4) {
        v2f a; a.x = arp[k];            a.y = arp[k + 1];
        v2f b; b.x = wp[(long)k * N];   b.y = wp[(long)(k + 1) * N];
        acc = __builtin_amdgcn_wmma_f32_16x16x4_f32(
            false, a, false, b, (short)0, acc, false, false);
    }
}

// -------- fused GEMM: out = relu?( A0@W0 + A1@W1 + A2@W2 + bias ) --------------
// grid.x = nRows/16 ; block = (N/16) waves * 32
__global__ void gemm_fused_kernel(
    const float* __restrict__ A0, const float* __restrict__ W0,
    const float* __restrict__ A1, const float* __restrict__ W1,
    const float* __restrict__ A2, const float* __restrict__ W2,
    const float* __restrict__ bias, float* __restrict__ out,
    int nRows, int N, int doRelu)
{
    __shared__ float Alds[16 * LDSW];
    const int tid  = threadIdx.x;
    const int lane = tid & 31;
    const int wave = tid >> 5;
    const int rowBase = blockIdx.x * 16;
    const int nBase   = wave * 16;
    const int col  = lane & 15;
    const int half = lane >> 4;

    v8f acc;
    float bv = bias[nBase + col];
    #pragma unroll
    for (int i = 0; i < 8; ++i) acc[i] = bv;

    accum_src(A0, W0, 0, Alds, rowBase, nRows, N, nBase, tid, blockDim.x, lane, acc);
    if (A1) accum_src(A1, W1, 1, Alds, rowBase, nRows, N, nBase, tid, blockDim.x, lane, acc);
    if (A2) accum_src(A2, W2, 1, Alds, rowBase, nRows, N, nBase, tid, blockDim.x, lane, acc);

    if (doRelu) {
        #pragma unroll
        for (int i = 0; i < 8; ++i) acc[i] = fmaxf(acc[i], 0.f);
    }

    // C/D layout: VGPR r -> row (r + 8*half), column col
    #pragma unroll
    for (int r = 0; r < 8; ++r) {
        int grow = rowBase + r + half * 8;
        if (grow < nRows) out[(long)grow * N + nBase + col] = acc[r];
    }
}

extern "C" void kernel_launch(void* const* d_in, const int* in_sizes, int n_in,
                              void* d_out, int out_size, void* d_ws, size_t ws_size,
                              hipStream_t stream) {
    (void)n_in; (void)out_size; (void)ws_size;
    const float* xp = (const float*)d_in[0];
    const float* xa = (const float*)d_in[1];
    const int* wr_src = (const int*)d_in[2];
    const int* wr_dst = (const int*)d_in[3];
    const int* ci_src = (const int*)d_in[4];
    const int* ci_dst = (const int*)d_in[5];
    const int* wb_src = (const int*)d_in[6];
    const int* wb_dst = (const int*)d_in[7];
    const float* w1_w  = (const float*)d_in[8];   // 128x128
    const float* w1_c  = (const float*)d_in[9];
    const float* w1_wb = (const float*)d_in[10];
    const float* w1_rp = (const float*)d_in[11];
    const float* b1_rp = (const float*)d_in[12];
    const float* w1_ra = (const float*)d_in[13];
    const float* b1_ra = (const float*)d_in[14];
    const float* w2_w  = (const float*)d_in[15];  // 128x64
    const float* w2_c  = (const float*)d_in[16];
    const float* w2_wb = (const float*)d_in[17];
    const float* w2_rp = (const float*)d_in[18];
    const float* b2_rp = (const float*)d_in[19];
    const float* w2_ra = (const float*)d_in[20];
    const float* b2_ra = (const float*)d_in[21];
    const int E = in_sizes[2];

    // workspace layout (floats)
    float* ws   = (float*)d_ws;
    float* hp   = ws;                              // 100000*128
    float* ha   = hp  + (long)NPAPER * DFEAT;      //  50000*128
    float* aggW = ha  + (long)NAUTHOR * DFEAT;     // 100000*128 (writes -> paper)
    float* aggC = aggW + (long)NPAPER * DFEAT;     // 100000*128 (cites  -> paper)
    float* aggA = aggC + (long)NPAPER * DFEAT;     //  50000*128 (wb     -> author)
    const long aggFloats = ((long)NPAPER + NPAPER + NAUTHOR) * DFEAT;  // contiguous
    const long agg4 = aggFloats / 4;

    float* outP = (float*)d_out;                   // 100000*64
    float* outA = outP + (long)NPAPER * DOUT;      //  50000*64

    const int segBlocks  = (int)(((long)E * 32 + 255) / 256);
    const int fillBlocks = (int)((agg4 + 255) / 256);

    // ---- layer 1 aggregation ----
    fill_neginf_kernel<<<fillBlocks, 256, 0, stream>>>((float4*)aggW, agg4);
    segmax_kernel<<<segBlocks, 256, 0, stream>>>(xa, wr_src, wr_dst, aggW, E);
    segmax_kernel<<<segBlocks, 256, 0, stream>>>(xp, ci_src, ci_dst, aggC, E);
    segmax_kernel<<<segBlocks, 256, 0, stream>>>(xp, wb_src, wb_dst, aggA, E);

    // ---- layer 1 GEMMs (+bias, +relu) ----
    gemm_fused_kernel<<<NPAPER / 16, 256, 0, stream>>>(
        xp, w1_rp, aggW, w1_w, aggC, w1_c, b1_rp, hp, NPAPER, DFEAT, 1);
    gemm_fused_kernel<<<NAUTHOR / 16, 256, 0, stream>>>(
        xa, w1_ra, aggA, w1_wb, nullptr, nullptr, b1_ra, ha, NAUTHOR, DFEAT, 1);

    // ---- layer 2 aggregation ----
    fill_neginf_kernel<<<fillBlocks, 256, 0, stream>>>((float4*)aggW, agg4);
    segmax_kernel<<<segBlocks, 256, 0, stream>>>(ha, wr_src, wr_dst, aggW, E);
    segmax_kernel<<<segBlocks, 256, 0, stream>>>(hp, ci_src, ci_dst, aggC, E);
    segmax_kernel<<<segBlocks, 256, 0, stream>>>(hp, wb_src, wb_dst, aggA, E);

    // ---- layer 2 GEMMs (+bias) -> d_out ----
    gemm_fused_kernel<<<NPAPER / 16, 128, 0, stream>>>(
        hp, w2_rp, aggW, w2_w, aggC, w2_c, b2_rp, outP, NPAPER, DOUT, 0);
    gemm_fused_kernel<<<NAUTHOR / 16, 128, 0, stream>>>(
        ha, w2_ra, aggA, w2_wb, nullptr, nullptr, b2_ra, outA, NAUTHOR, DOUT, 0);
}